// GCNLayer_13357348290889
// MI455X (gfx1250) — compile-verified
//
#include <hip/hip_runtime.h>

// Problem constants (match reference)
#define B_    4
#define N_    50000
#define CIN_  64
#define E_    800000
#define HALF_ 32
#define M_    (B_ * N_)          // 200000 flattened rows, = 12500 * 16

typedef __attribute__((ext_vector_type(2))) float v2f;
typedef __attribute__((ext_vector_type(8))) float v8f;

// ---------------------------------------------------------------------------
// Kernel 1: fused GEMM  y[m, 0:64] = x[m, 0:64] @ Wc^T   (Wc = [W_lin; W_eye])
//   cols  0:32 -> z (workspace, pre-bias)        [feeds the SpMM scatter]
//   cols 32:64 -> out[..., 32:64] + b_eye        [eye branch, final]
//   also writes out[..., 0:32] = b_lin           [accumulator init for scatter]
// One wave computes one 16x16 output tile via 16x V_WMMA_F32_16X16X4_F32.
// 8 waves / block = 2 row-tiles x 4 col-tiles.
// ---------------------------------------------------------------------------
__global__ __launch_bounds__(256) void gcn_gemm_wmma(
    const float* __restrict__ x,
    const float* __restrict__ W_lin, const float* __restrict__ b_lin,
    const float* __restrict__ W_eye, const float* __restrict__ b_eye,
    float* __restrict__ out, float* __restrict__ z)
{
    // LDS: Wc[n][k], n = output channel (0..63), k = input channel (0..63)
    __shared__ float lw[64 * 64];   // 16 KB

    const int tid = threadIdx.x;
    #pragma unroll
    for (int j = 0; j < 16; ++j) {
        int idx = tid + 256 * j;          // 4096 total
        int n = idx >> 6, k = idx & 63;
        lw[idx] = (n < HALF_) ? W_lin[n * CIN_ + k]
                              : W_eye[(n - HALF_) * CIN_ + k];
    }
    __syncthreads();

    const int wave  = tid >> 5;
    const int lane  = tid & 31;
    const int hi    = lane >> 4;      // half-wave: K-pair select
    const int l16   = lane & 15;
    const int mtile = blockIdx.x * 2 + (wave >> 2);
    const int nt    = wave & 3;       // 16-wide column tile (0..3)

    const int row  = mtile * 16 + l16;   // A-matrix row held by this lane
    const int ncol = nt * 16 + l16;      // B-matrix column held by this lane

    // 16x4 f32 A layout: lanes 0-15 -> K={0,1}; lanes 16-31 -> K={2,3}
    const float* xrow = x  + (long)row * CIN_ + hi * 2;
    const float* wcol = lw + ncol * CIN_ + hi * 2;

    v8f acc = {};
    #pragma unroll
    for (int kb = 0; kb < 16; ++kb) {
        v2f a = *(const v2f*)(xrow + kb * 4);   // global_load_b64
        v2f b = *(const v2f*)(wcol + kb * 4);   // ds_load_b64
        acc = __builtin_amdgcn_wmma_f32_16x16x4_f32(
            /*neg_a=*/false, a, /*neg_b=*/false, b,
            /*c_mod=*/(short)0, acc, /*reuse_a=*/false, /*reuse_b=*/false);
    }

    // D layout: VGPR r holds M = r + 8*hi, N = l16 (within tile)
    const int m0 = mtile * 16 + 8 * hi;
    if (nt < 2) {
        // lin half: z = x @ W_lin^T (pre-bias); init out accumulator with b_lin
        const float bl = b_lin[ncol];            // ncol in [0,32)
        #pragma unroll
        for (int r = 0; r < 8; ++r) {
            const long m = m0 + r;
            z[m * HALF_ + ncol]  = acc[r];
            out[m * CIN_ + ncol] = bl;
        }
    } else {
        // eye half: final value straight into out[..., 32:64]
        const float be = b_eye[ncol - HALF_];    // ncol in [32,64) == out col
        #pragma unroll
        for (int r = 0; r < 8; ++r) {
            const long m = m0 + r;
            out[m * CIN_ + ncol] = acc[r] + be;
        }
    }
}

// ---------------------------------------------------------------------------
// Kernel 2: edge scatter   out[b, rows[e], c] += vals[e] * z[b, cols[e], c]
// 128 threads per edge (4 batches x 32 channels). Each wave: one (e,b) pair,
// lanes cover c=0..31 -> coalesced z gather + coalesced f32 atomics.
// Hardware atomic forced via inline asm (device scope: RMW performed at L2,
// coherent across all WGPs) to avoid any CAS-loop fallback on the hot path.
// ---------------------------------------------------------------------------
__global__ __launch_bounds__(256) void gcn_edge_scatter(
    const float* __restrict__ z, const float* __restrict__ vals,
    const int* __restrict__ rows, const int* __restrict__ cols,
    float* __restrict__ out)
{
    const long t = (long)blockIdx.x * 256 + threadIdx.x;
    const int e = (int)(t >> 7);        // edge
    const int r = (int)(t & 127);
    const int b = r >> 5;               // batch
    const int c = r & 31;               // channel

    const int   cn = cols[e];           // wave-uniform (broadcast load)
    const int   rn = rows[e];
    const float v  = vals[e];

    const float zv  = z[((long)b * N_ + cn) * HALF_ + c];
    const float upd = v * zv;
    float* dst = out + ((long)b * N_ + rn) * CIN_ + c;

    // Non-returning f32 atomic add, device scope (STOREcnt-tracked)
    asm volatile("global_atomic_add_f32 %0, %1, off scope:SCOPE_DEV"
                 :
                 : "v"(dst), "v"(upd)
                 : "memory");
}

// ---------------------------------------------------------------------------
extern "C" void kernel_launch(void* const* d_in, const int* in_sizes, int n_in,
                              void* d_out, int out_size, void* d_ws, size_t ws_size,
                              hipStream_t stream) {
    const float* x     = (const float*)d_in[0];
    const float* vals  = (const float*)d_in[1];
    const float* W_lin = (const float*)d_in[2];
    const float* b_lin = (const float*)d_in[3];
    const float* W_eye = (const float*)d_in[4];
    const float* b_eye = (const float*)d_in[5];
    const int*   rows  = (const int*)d_in[6];
    const int*   cols  = (const int*)d_in[7];
    float* out = (float*)d_out;
    float* z   = (float*)d_ws;          // needs M_*HALF_*4 = 25.6 MB

    // 12500 row-tiles of 16, 2 per block -> 6250 blocks
    gcn_gemm_wmma<<<M_ / 32, 256, 0, stream>>>(x, W_lin, b_lin, W_eye, b_eye, out, z);

    // E*B*32 = 102.4M threads, exact multiple of 256 -> 400000 blocks
    gcn_edge_scatter<<<(long)E_ * 128 / 256, 256, 0, stream>>>(z, vals, rows, cols, out);
}